// InstanceClusteringModule_38259568672933
// MI455X (gfx1250) — compile-verified
//
#include <hip/hip_runtime.h>

typedef __attribute__((ext_vector_type(2))) float v2f;
typedef __attribute__((ext_vector_type(8))) float v8f;
typedef __attribute__((ext_vector_type(4))) unsigned int v4u;
typedef __attribute__((ext_vector_type(8))) int v8i;
typedef __attribute__((ext_vector_type(4))) int v4i;

#define N_PTS 100000
#define DIM   1024
#define KC    64

#define CHUNK   32            // columns per LDS chunk
#define FSTRIDE 36            // 32 cols + 4 dword pad -> conflict-free fragment reads
#define NCHUNK  (DIM / CHUNK)

#if defined(__has_builtin)
#if __has_builtin(__builtin_amdgcn_tensor_load_to_lds)
#define HAVE_TDM 1
#endif
#if __has_builtin(__builtin_amdgcn_s_wait_tensorcnt)
#define HAVE_WAIT_TENSOR 1
#endif
#endif

__device__ __forceinline__ void wait_tensor0() {
#ifdef HAVE_WAIT_TENSOR
  __builtin_amdgcn_s_wait_tensorcnt(0);
#else
  asm volatile("s_wait_tensorcnt 0x0" ::: "memory");
#endif
}

// Issue one TDM 2D tile load: global (tensor_d1 rows x tensor_d0 cols, row
// stride stride0 elems, 4B elems) -> LDS tile (tile_d1 x tile_d0) with
// hardware row padding (pad after 2^(pil+1) dwords, pad (pam+1) dwords).
__device__ __forceinline__ void tdm_load_2d(unsigned int lds_addr, const float* gptr,
                                            unsigned int tensor_d0, unsigned int tensor_d1,
                                            unsigned int tile_d0, unsigned int tile_d1,
                                            unsigned int stride0,
                                            unsigned int pil, unsigned int pam) {
#ifdef HAVE_TDM
  unsigned long long ga = (unsigned long long)(uintptr_t)gptr;
  v4u g0;
  g0.x = 1u;                                               // count=1 (valid), no gather
  g0.y = lds_addr;                                         // LDS byte address
  g0.z = (unsigned int)(ga & 0xffffffffu);                 // global_addr[31:0]
  g0.w = (unsigned int)((ga >> 32) & 0x01ffffffu) | 0x80000000u; // ga[56:32] | type=2
  v8i g1;
  g1[0] = (int)((2u << 16)            // data_size = 4B
                | (1u << 20)          // pad_enable
                | (pil << 22)         // pad_interval
                | (pam << 25));       // pad_amount
  g1[1] = (int)((tensor_d0 & 0xffffu) << 16);                      // dim0[15:0]
  g1[2] = (int)((tensor_d0 >> 16) | ((tensor_d1 & 0xffffu) << 16));// dim0[31:16]|dim1[15:0]
  g1[3] = (int)((tensor_d1 >> 16) | (tile_d0 << 16));              // dim1[31:16]|tile0
  g1[4] = (int)tile_d1;                                            // tile1 (tile2=0)
  g1[5] = (int)stride0;                                            // dim0_stride[31:0]
  g1[6] = 0;
  g1[7] = 0;
  v4i gz = (v4i){0, 0, 0, 0};
#if __clang_major__ >= 23
  v8i gz8 = (v8i){0, 0, 0, 0, 0, 0, 0, 0};
  __builtin_amdgcn_tensor_load_to_lds(g0, g1, gz, gz, gz8, 0);
#else
  __builtin_amdgcn_tensor_load_to_lds(g0, g1, gz, gz, 0);
#endif
#endif
}

// ---------------- ws layout (bytes) ----------------
// [0,256)              : c2  (float[64])
// [256,512)            : counts (int[64])
// [512,512+256K)       : sums (float[64*1024])
// [512+256K, +400000)  : ids  (int[100000])

// K0: per-cluster squared norms
__global__ __launch_bounds__(256) void k_c2(const float* __restrict__ C,
                                            float* __restrict__ c2) {
  __shared__ float red[256];
  const int k = blockIdx.x;
  float s = 0.f;
  for (int d = threadIdx.x; d < DIM; d += 256) {
    float v = C[(size_t)k * DIM + d];
    s += v * v;
  }
  red[threadIdx.x] = s;
  __syncthreads();
  for (int off = 128; off > 0; off >>= 1) {
    if (threadIdx.x < (unsigned)off) red[threadIdx.x] += red[threadIdx.x + off];
    __syncthreads();
  }
  if (threadIdx.x == 0) c2[k] = red[0];
}

// K1: TDM-fed, double-buffered FP32 WMMA distance GEMM + in-register argmin.
__global__ __launch_bounds__(256) void k_assign(const float* __restrict__ F,
                                                const float* __restrict__ C,
                                                const float* __restrict__ c2,
                                                int* __restrict__ counts,
                                                int* __restrict__ ids,
                                                float* __restrict__ ids_f) {
  __shared__ float sF[2 * 128 * FSTRIDE];  // double-buffered feature tiles
  __shared__ float sC[2 * 64 * FSTRIDE];   // double-buffered center tiles
  __shared__ float sc2[64];

  const int tid = threadIdx.x;
  const int l = tid & 31;
  const int w = tid >> 5;
  const int rowBase = blockIdx.x * 128;
  const unsigned int rowsRem = (unsigned int)(N_PTS - rowBase);  // >0 by grid sizing

  if (tid < 64) sc2[tid] = c2[tid];

  const int lm = l & 15;
  const int lh = l >> 4;
  const int fragOff = lh * 2;

  v8f acc[4];
#pragma unroll
  for (int t = 0; t < 4; ++t) acc[t] = (v8f){0, 0, 0, 0, 0, 0, 0, 0};

#ifdef HAVE_TDM
  const unsigned int ldsF0 = (unsigned int)(uintptr_t)&sF[0];
  const unsigned int ldsF1 = (unsigned int)(uintptr_t)&sF[128 * FSTRIDE];
  const unsigned int ldsC0 = (unsigned int)(uintptr_t)&sC[0];
  const unsigned int ldsC1 = (unsigned int)(uintptr_t)&sC[64 * FSTRIDE];
  if (w == 0) {  // prologue: chunk 0 into buffer 0
    tdm_load_2d(ldsF0, F + (size_t)rowBase * DIM, DIM, rowsRem, CHUNK, 128, DIM, 4, 3);
    tdm_load_2d(ldsC0, C, DIM, KC, CHUNK, KC, DIM, 4, 3);
  }
#endif

  for (int c = 0; c < NCHUNK; ++c) {
    const int buf = c & 1;
    const float* bF = &sF[buf * 128 * FSTRIDE];
    const float* bC = &sC[buf * 64 * FSTRIDE];

#ifdef HAVE_TDM
    if (w == 0) wait_tensor0();        // chunk c resident
    __syncthreads();                   // everyone done with buf^1 & sees chunk c
    if (w == 0 && c + 1 < NCHUNK) {    // overlap: DMA chunk c+1 into other buffer
      const int c0n = (c + 1) * CHUNK;
      tdm_load_2d(buf ? ldsF0 : ldsF1, F + (size_t)rowBase * DIM + c0n,
                  (unsigned int)(DIM - c0n), rowsRem, CHUNK, 128, DIM, 4, 3);
      tdm_load_2d(buf ? ldsC0 : ldsC1, C + c0n,
                  (unsigned int)(DIM - c0n), KC, CHUNK, KC, DIM, 4, 3);
    }
#else
    {  // fallback staging: batched loads -> LDS (single logical step per chunk)
      const int c0 = c * CHUNK;
      float tF[16];
#pragma unroll
      for (int i = 0; i < 16; ++i) {
        int rg = rowBase + w * 16 + i;
        if (rg >= N_PTS) rg = N_PTS - 1;
        tF[i] = F[(size_t)rg * DIM + c0 + l];
      }
#pragma unroll
      for (int i = 0; i < 16; ++i) ((float*)bF)[(w * 16 + i) * FSTRIDE + l] = tF[i];
      float tC[8];
#pragma unroll
      for (int i = 0; i < 8; ++i) tC[i] = C[(size_t)(w * 8 + i) * DIM + c0 + l];
#pragma unroll
      for (int i = 0; i < 8; ++i) ((float*)bC)[(w * 8 + i) * FSTRIDE + l] = tC[i];
      __syncthreads();
    }
#endif

    const int arow = (w * 16 + lm) * FSTRIDE;
#pragma unroll
    for (int d = 0; d < CHUNK; d += 4) {
      v2f a = *(const v2f*)&bF[arow + d + fragOff];
#pragma unroll
      for (int t = 0; t < 4; ++t) {
        v2f b = *(const v2f*)&bC[(t * 16 + lm) * FSTRIDE + d + fragOff];
        acc[t] = __builtin_amdgcn_wmma_f32_16x16x4_f32(
            false, a, false, b, (short)0, acc[t], false, false);
      }
    }
#ifndef HAVE_TDM
    __syncthreads();
#endif
  }
#ifdef HAVE_TDM
  __syncthreads();  // ensure sc2 visible / all compute done before readback below
#endif

  // argmin per row: score = c2[k] - 2*cross (f2 row-constant, dropped).
#pragma unroll
  for (int v = 0; v < 8; ++v) {
    float best = sc2[lm] - 2.0f * acc[0][v];
    int bi = lm;
#pragma unroll
    for (int t = 1; t < 4; ++t) {
      float val = sc2[t * 16 + lm] - 2.0f * acc[t][v];
      if (val < best) { best = val; bi = t * 16 + lm; }
    }
#pragma unroll
    for (int m = 1; m < 16; m <<= 1) {
      float ov = __shfl_xor(best, m, 32);
      int   oi = __shfl_xor(bi, m, 32);
      if (ov < best || (ov == best && oi < bi)) { best = ov; bi = oi; }
    }
    if (lm == v) {
      int rg = rowBase + w * 16 + lh * 8 + v;
      if (rg < N_PTS) {
        ids[rg] = bi;
        ids_f[rg] = (float)bi;
        atomicAdd(&counts[bi], 1);
      }
    }
  }
}

// K2: deterministic segment-sum. One block per (cluster, 128-col block).
__global__ __launch_bounds__(256) void k_segsum(const float* __restrict__ F,
                                                const int* __restrict__ ids,
                                                float* __restrict__ sums) {
  __shared__ float part[8 * 128];
  const int k  = blockIdx.x >> 3;
  const int cb = (blockIdx.x & 7) * 128;
  const int tid = threadIdx.x;
  const int l = tid & 31;
  const int w = tid >> 5;

  const int slice = (N_PTS + 7) / 8;  // 12500
  const int rs = w * slice;
  const int re = (rs + slice < N_PTS) ? (rs + slice) : N_PTS;

  float s0 = 0.f, s1 = 0.f, s2 = 0.f, s3 = 0.f;
  for (int base = rs; base < re; base += 32) {
    int r = base + l;
    bool m = (r < re) && (ids[r] == k);
    unsigned long long mask = __ballot(m);
    while (mask) {  // fixed ascending order -> deterministic
      int j = __ffsll(mask) - 1;
      mask &= mask - 1;
      const float4 f = *(const float4*)&F[(size_t)(base + j) * DIM + cb + l * 4];
      s0 += f.x; s1 += f.y; s2 += f.z; s3 += f.w;
    }
  }
  part[w * 128 + l * 4 + 0] = s0;
  part[w * 128 + l * 4 + 1] = s1;
  part[w * 128 + l * 4 + 2] = s2;
  part[w * 128 + l * 4 + 3] = s3;
  __syncthreads();
  if (tid < 128) {
    float s = 0.f;
#pragma unroll
    for (int ww = 0; ww < 8; ++ww) s += part[ww * 128 + tid];  // fixed order
    sums[(size_t)k * DIM + cb + tid] = s;
  }
}

// K3: means with empty-cluster fallback
__global__ __launch_bounds__(256) void k_final(const float* __restrict__ sums,
                                               const int* __restrict__ counts,
                                               const float* __restrict__ C,
                                               float* __restrict__ out) {
  int i = blockIdx.x * 256 + threadIdx.x;
  if (i < KC * DIM) {
    int k = i >> 10;
    int c = counts[k];
    out[i] = (c > 0) ? sums[i] / (float)c : C[i];
  }
}

extern "C" void kernel_launch(void* const* d_in, const int* in_sizes, int n_in,
                              void* d_out, int out_size, void* d_ws, size_t ws_size,
                              hipStream_t stream) {
  const float* F = (const float*)d_in[0];  // (N, D) float32
  const float* C = (const float*)d_in[1];  // (K, D) float32
  float* out = (float*)d_out;              // [K*D means | N ids]

  char* ws = (char*)d_ws;
  float* c2     = (float*)(ws);
  int*   counts = (int*)(ws + 256);
  float* sums   = (float*)(ws + 512);
  int*   ids    = (int*)(ws + 512 + (size_t)KC * DIM * sizeof(float));

  hipMemsetAsync(counts, 0, KC * sizeof(int), stream);
  k_c2<<<KC, 256, 0, stream>>>(C, c2);
  k_assign<<<(N_PTS + 127) / 128, 256, 0, stream>>>(F, C, c2, counts, ids,
                                                    out + (size_t)KC * DIM);
  k_segsum<<<KC * 8, 256, 0, stream>>>(F, ids, sums);
  k_final<<<(KC * DIM + 255) / 256, 256, 0, stream>>>(sums, counts, C, out);
}